// InteractionNetTaggerEmbedding_56977036149126
// MI455X (gfx1250) — compile-verified
//
#include <hip/hip_runtime.h>
#include <hip/hip_bf16.h>

// InteractionNetTagger — CDNA5 (gfx1250) fp32 WMMA implementation, round 2.
// One 256-thread block (8 wave32 waves) per batch element.
// All GEMMs via V_WMMA_F32_16X16X4_F32 (exact fp32).
// - fr layer 1 factored: preact(r,s) = U[r] + V[s] + b1  (60-row GEMM)
// - all LDS reads unconditional (padded buffers + dummy "-inf" sender row)
// - A-frags and B-frags loaded as single ds_load_b64 (k-pair-interleaved W)
// - fr_w2/fr_w3 B-fragments register-cached across the 60-receiver loop
// - h2 double-buffered: one barrier per receiver

#define NPART 60
#define PFEAT 30
#define DE_   20
#define DO_   24
#define HID_  60

typedef __attribute__((ext_vector_type(2))) float v2f;
typedef __attribute__((ext_vector_type(8))) float v8f;

__device__ __forceinline__ v8f wmma4(v2f a, v2f b, v8f c) {
  return __builtin_amdgcn_wmma_f32_16x16x4_f32(false, a, false, b, (short)0, c,
                                               false, false);
}

// ---- LDS layout (float offsets; all even for b64 alignment) ---------------
// persistent:
#define OFF_EBAR 0              // 60*20; reused as 24-wide output accumulator
#define SB       1216
// phase B/C:
#define OFF_XT   (SB + 0)       // [64][32]  xT, rows>=60 & cols>=30 zero
#define OFF_P    (SB + 2048)    // [65][120] rows 0..59 = [U+b1 | V]; row 64 = -1e30
#define OFF_W1P  (SB + 9848)    // 16 kpairs x 128 x v2f  (fused [w1_top|w1_bot])
#define OFF_W2P  (SB + 13944)   // 30 kpairs x 64  x v2f  (fr_w2, cols>=60 zero)
#define OFF_W3P  (SB + 17784)   // 30 kpairs x 32  x v2f  (fr_w3, cols>=20 zero)
#define OFF_H2A  (SB + 19704)   // [64][64]
#define OFF_H2B  (SB + 23800)   // [64][64]
#define LDS_FLOATS (SB + 27896)
// phase D (aliases SB..):
#define OFF_CMAT  (SB + 0)      // [64][56] [x^T | Ebar | 0]
#define OFF_FOW1P (SB + 3584)   // 26 kpairs x 64 x v2f
#define OFF_HD    (SB + 6912)   // [64][64]
#define OFF_FOW2P (SB + 11008)  // 30 kpairs x 64 x v2f
#define OFF_HD2   (SB + 14848)  // [64][64]
#define OFF_FOW3P (SB + 18944)  // 30 kpairs x 32 x v2f

__global__ __launch_bounds__(256) void intnet_tagger_kernel(
    const float* __restrict__ x,
    const float* __restrict__ fr_w1, const float* __restrict__ fr_b1,
    const float* __restrict__ fr_w2, const float* __restrict__ fr_b2,
    const float* __restrict__ fr_w3, const float* __restrict__ fr_b3,
    const float* __restrict__ fo_w1, const float* __restrict__ fo_b1,
    const float* __restrict__ fo_w2, const float* __restrict__ fo_b2,
    const float* __restrict__ fo_w3, const float* __restrict__ fo_b3,
    float* __restrict__ out)
{
  extern __shared__ float sm[];
  const int b    = blockIdx.x;
  const int tid  = threadIdx.x;
  const int wid  = tid >> 5;
  const int lane = tid & 31;
  const int l15  = lane & 15;
  const int khi  = (lane >> 4) << 1;  // A/B frag: hi half-lanes hold K+2,K+3
  const int hk   = lane >> 4;         // which element of a k-pair-pair
  const int mHf  = (lane >> 4) << 3;  // D frag: hi half-lanes hold rows M+8..

  const float* xb = x + (size_t)b * (PFEAT * NPART);

  // ---- Stage A: cooperative LDS fills -------------------------------------
  for (int i = tid; i < 64 * 32; i += 256) {           // xT[p][f]
    int p = i >> 5, f = i & 31;
    sm[OFF_XT + i] = (p < NPART && f < PFEAT) ? xb[f * NPART + p] : 0.0f;
  }
  for (int i = tid; i < 120; i += 256)                 // dummy sender row
    sm[OFF_P + 64 * 120 + i] = -1.0e30f;
  for (int i = tid; i < 16 * 128; i += 256) {          // fused w1, pair layout
    int kp = i >> 7, n = i & 127;
    int k0 = 2 * kp, k1 = 2 * kp + 1;
    float v0 = 0.0f, v1 = 0.0f;
    if (n < 120) {
      const int c  = (n < HID_) ? n : (n - HID_);
      const int ro = (n < HID_) ? 0 : PFEAT;
      if (k0 < PFEAT) v0 = fr_w1[(ro + k0) * HID_ + c];
      if (k1 < PFEAT) v1 = fr_w1[(ro + k1) * HID_ + c];
    }
    sm[OFF_W1P + 2 * i]     = v0;
    sm[OFF_W1P + 2 * i + 1] = v1;
  }
  for (int i = tid; i < 30 * 64; i += 256) {           // fr_w2, pair layout
    int kp = i >> 6, n = i & 63;
    float v0 = 0.0f, v1 = 0.0f;
    if (n < HID_) { v0 = fr_w2[(2 * kp) * HID_ + n]; v1 = fr_w2[(2 * kp + 1) * HID_ + n]; }
    sm[OFF_W2P + 2 * i] = v0; sm[OFF_W2P + 2 * i + 1] = v1;
  }
  for (int i = tid; i < 30 * 32; i += 256) {           // fr_w3, pair layout
    int kp = i >> 5, n = i & 31;
    float v0 = 0.0f, v1 = 0.0f;
    if (n < DE_) { v0 = fr_w3[(2 * kp) * DE_ + n]; v1 = fr_w3[(2 * kp + 1) * DE_ + n]; }
    sm[OFF_W3P + 2 * i] = v0; sm[OFF_W3P + 2 * i + 1] = v1;
  }
  for (int i = tid; i < NPART * DE_; i += 256) sm[OFF_EBAR + i] = 0.0f;
  __syncthreads();

  // ---- Stage B: P = [xT @ w1_top + b1 | xT @ w1_bot] ----------------------
  // 32 output tiles (4 Mt x 8 Nt); each wave owns one Nt column, 4 Mt tiles.
  {
    const int nB = wid * 16 + l15;  // 0..127
    v8f a0 = {}, a1 = {}, a2 = {}, a3 = {};
#pragma unroll
    for (int kp = 0; kp < 8; ++kp) {
      const int kk = kp * 4 + khi;
      const v2f bf = *(const v2f*)&sm[OFF_W1P + ((kp * 2 + hk) * 128 + nB) * 2];
      const v2f x0 = *(const v2f*)&sm[OFF_XT + (0 * 16 + l15) * 32 + kk];
      const v2f x1 = *(const v2f*)&sm[OFF_XT + (1 * 16 + l15) * 32 + kk];
      const v2f x2 = *(const v2f*)&sm[OFF_XT + (2 * 16 + l15) * 32 + kk];
      const v2f x3 = *(const v2f*)&sm[OFF_XT + (3 * 16 + l15) * 32 + kk];
      a0 = wmma4(x0, bf, a0);
      a1 = wmma4(x1, bf, a1);
      a2 = wmma4(x2, bf, a2);
      a3 = wmma4(x3, bf, a3);
    }
    if (nB < 120) {
      const float b1v = (nB < HID_) ? fr_b1[nB] : 0.0f;
#pragma unroll
      for (int v = 0; v < 8; ++v) {
        sm[OFF_P + (0 * 16 + mHf + v) * 120 + nB] = a0[v] + b1v;
        sm[OFF_P + (1 * 16 + mHf + v) * 120 + nB] = a1[v] + b1v;
        sm[OFF_P + (2 * 16 + mHf + v) * 120 + nB] = a2[v] + b1v;
        sm[OFF_P + (3 * 16 + mHf + v) * 120 + nB] = a3[v] + b1v;
      }
    }
  }
  __syncthreads();

  // ---- Stage C: per-receiver fr layers 2,3 + scatter-add ------------------
  const int n2 = (wid & 3) * 16 + l15;   // layer-2 column
  const int n3 = (wid & 1) * 16 + l15;   // layer-3 column
  const float b2v = (n2 < HID_) ? fr_b2[n2] : 0.0f;
  const float b3v = (n3 < DE_)  ? fr_b3[n3] : 0.0f;

  v2f w2f[15], w3f[15];                  // B-frags invariant across receivers
#pragma unroll
  for (int kp = 0; kp < 15; ++kp) {
    w2f[kp] = *(const v2f*)&sm[OFF_W2P + ((kp * 2 + hk) * 64 + n2) * 2];
    w3f[kp] = *(const v2f*)&sm[OFF_W3P + ((kp * 2 + hk) * 32 + n3) * 2];
  }

  for (int r = 0; r < NPART; ++r) {
    float* h2w = &sm[(r & 1) ? OFF_H2B : OFF_H2A];

    // layer 2: h2 = relu(h1 @ fr_w2 + b2); h1 rows generated from P on the fly
    {
      const int MtA = wid >> 2;              // tiles MtA and MtA+2 (same Nt)
      const int m0 = MtA * 16 + l15;         // 0..31  (always a valid sender)
      const int m1 = m0 + 32;                // 32..63 (row>=59 -> dummy row 64)
      const int s0 = (m0 < r) ? m0 : m0 + 1;
      const int s1 = (m1 < NPART - 1) ? ((m1 < r) ? m1 : m1 + 1) : 64;
      const float* Pr  = &sm[OFF_P + r  * 120];
      const float* Ps0 = &sm[OFF_P + s0 * 120 + HID_];
      const float* Ps1 = &sm[OFF_P + s1 * 120 + HID_];
      v8f acc0 = {}, acc1 = {};
#pragma unroll
      for (int kp = 0; kp < 15; ++kp) {
        const int kk = kp * 4 + khi;
        const v2f pr  = *(const v2f*)&Pr[kk];
        const v2f ps0 = *(const v2f*)&Ps0[kk];
        const v2f ps1 = *(const v2f*)&Ps1[kk];
        v2f a0, a1;
        a0.x = fmaxf(pr.x + ps0.x, 0.0f); a0.y = fmaxf(pr.y + ps0.y, 0.0f);
        a1.x = fmaxf(pr.x + ps1.x, 0.0f); a1.y = fmaxf(pr.y + ps1.y, 0.0f);
        acc0 = wmma4(a0, w2f[kp], acc0);
        acc1 = wmma4(a1, w2f[kp], acc1);
      }
#pragma unroll
      for (int v = 0; v < 8; ++v) {
        h2w[(MtA * 16      + mHf + v) * 64 + n2] = fmaxf(acc0[v] + b2v, 0.0f);
        h2w[(MtA * 16 + 32 + mHf + v) * 64 + n2] = fmaxf(acc1[v] + b2v, 0.0f);
      }
    }
    __syncthreads();

    // layer 3: E = relu(h2 @ fr_w3 + b3); column-sum valid rows into Ebar[r]
    {
      const int Mt3 = wid >> 1;
      const int m3  = Mt3 * 16 + l15;
      v8f acc = {};
#pragma unroll
      for (int kp = 0; kp < 15; ++kp) {
        const int kk = kp * 4 + khi;
        const v2f a = *(const v2f*)&h2w[m3 * 64 + kk];
        acc = wmma4(a, w3f[kp], acc);
      }
      if (n3 < DE_) {
        float ssum = 0.0f;
#pragma unroll
        for (int v = 0; v < 8; ++v) {
          const int mm = Mt3 * 16 + mHf + v;
          if (mm < NPART - 1) ssum += fmaxf(acc[v] + b3v, 0.0f);
        }
        atomicAdd(&sm[OFF_EBAR + r * DE_ + n3], ssum);  // ds_add_f32
      }
    }
    // no second barrier: next receiver writes the other h2 buffer
  }
  __syncthreads();

  // ---- Stage D: fo-MLP on particles + particle sum ------------------------
  for (int i = tid; i < 64 * 56; i += 256) {           // Cmat = [x^T | Ebar | 0]
    int p = i / 56, n = i % 56;
    float v = 0.0f;
    if (p < NPART) {
      if (n < PFEAT)            v = xb[n * NPART + p];
      else if (n < PFEAT + DE_) v = sm[OFF_EBAR + p * DE_ + (n - PFEAT)];
    }
    sm[OFF_CMAT + i] = v;
  }
  for (int i = tid; i < 26 * 64; i += 256) {           // fo_w1, pair layout
    int kp = i >> 6, n = i & 63;
    int k0 = 2 * kp, k1 = 2 * kp + 1;
    float v0 = 0.0f, v1 = 0.0f;
    if (n < HID_) {
      if (k0 < PFEAT + DE_) v0 = fo_w1[k0 * HID_ + n];
      if (k1 < PFEAT + DE_) v1 = fo_w1[k1 * HID_ + n];
    }
    sm[OFF_FOW1P + 2 * i] = v0; sm[OFF_FOW1P + 2 * i + 1] = v1;
  }
  for (int i = tid; i < 30 * 64; i += 256) {           // fo_w2, pair layout
    int kp = i >> 6, n = i & 63;
    float v0 = 0.0f, v1 = 0.0f;
    if (n < HID_) { v0 = fo_w2[(2 * kp) * HID_ + n]; v1 = fo_w2[(2 * kp + 1) * HID_ + n]; }
    sm[OFF_FOW2P + 2 * i] = v0; sm[OFF_FOW2P + 2 * i + 1] = v1;
  }
  for (int i = tid; i < 30 * 32; i += 256) {           // fo_w3, pair layout
    int kp = i >> 5, n = i & 31;
    float v0 = 0.0f, v1 = 0.0f;
    if (n < DO_) { v0 = fo_w3[(2 * kp) * DO_ + n]; v1 = fo_w3[(2 * kp + 1) * DO_ + n]; }
    sm[OFF_FOW3P + 2 * i] = v0; sm[OFF_FOW3P + 2 * i + 1] = v1;
  }
  __syncthreads();

  // fo layer 1: hD = relu(Cmat @ fo_w1 + b1), K=50 (padded to 52)
  {
    const int Nt  = wid & 3;
    const int MtA = wid >> 2;
    const int nD  = Nt * 16 + l15;
    const int m0  = MtA * 16 + l15;
    v8f acc0 = {}, acc1 = {};
#pragma unroll
    for (int kp = 0; kp < 13; ++kp) {
      const int kk = kp * 4 + khi;
      const v2f bf = *(const v2f*)&sm[OFF_FOW1P + ((kp * 2 + hk) * 64 + nD) * 2];
      const v2f a0 = *(const v2f*)&sm[OFF_CMAT + m0 * 56 + kk];
      const v2f a1 = *(const v2f*)&sm[OFF_CMAT + (m0 + 32) * 56 + kk];
      acc0 = wmma4(a0, bf, acc0);
      acc1 = wmma4(a1, bf, acc1);
    }
    const float bv = (nD < HID_) ? fo_b1[nD] : 0.0f;
#pragma unroll
    for (int v = 0; v < 8; ++v) {
      sm[OFF_HD + (MtA * 16      + mHf + v) * 64 + nD] = fmaxf(acc0[v] + bv, 0.0f);
      sm[OFF_HD + (MtA * 16 + 32 + mHf + v) * 64 + nD] = fmaxf(acc1[v] + bv, 0.0f);
    }
  }
  if (tid < DO_) sm[OFF_EBAR + tid] = 0.0f;  // Ebar reads done; reuse as out-acc
  __syncthreads();

  // fo layer 2: hD2 = relu(hD @ fo_w2 + b2)
  {
    const int Nt  = wid & 3;
    const int MtA = wid >> 2;
    const int nD  = Nt * 16 + l15;
    const int m0  = MtA * 16 + l15;
    v8f acc0 = {}, acc1 = {};
#pragma unroll
    for (int kp = 0; kp < 15; ++kp) {
      const int kk = kp * 4 + khi;
      const v2f bf = *(const v2f*)&sm[OFF_FOW2P + ((kp * 2 + hk) * 64 + nD) * 2];
      const v2f a0 = *(const v2f*)&sm[OFF_HD + m0 * 64 + kk];
      const v2f a1 = *(const v2f*)&sm[OFF_HD + (m0 + 32) * 64 + kk];
      acc0 = wmma4(a0, bf, acc0);
      acc1 = wmma4(a1, bf, acc1);
    }
    const float bv = (nD < HID_) ? fo_b2[nD] : 0.0f;
#pragma unroll
    for (int v = 0; v < 8; ++v) {
      sm[OFF_HD2 + (MtA * 16      + mHf + v) * 64 + nD] = fmaxf(acc0[v] + bv, 0.0f);
      sm[OFF_HD2 + (MtA * 16 + 32 + mHf + v) * 64 + nD] = fmaxf(acc1[v] + bv, 0.0f);
    }
  }
  __syncthreads();

  // fo layer 3 + particle sum: out[b] = sum_p relu(hD2 @ fo_w3 + b3)
  {
    const int Mt = wid >> 1;
    const int nD = (wid & 1) * 16 + l15;
    const int m  = Mt * 16 + l15;
    v8f acc = {};
#pragma unroll
    for (int kp = 0; kp < 15; ++kp) {
      const int kk = kp * 4 + khi;
      const v2f a  = *(const v2f*)&sm[OFF_HD2 + m * 64 + kk];
      const v2f bf = *(const v2f*)&sm[OFF_FOW3P + ((kp * 2 + hk) * 32 + nD) * 2];
      acc = wmma4(a, bf, acc);
    }
    if (nD < DO_) {
      const float bv = fo_b3[nD];
      float ssum = 0.0f;
#pragma unroll
      for (int v = 0; v < 8; ++v) {
        const int mm = Mt * 16 + mHf + v;
        if (mm < NPART) ssum += fmaxf(acc[v] + bv, 0.0f);
      }
      atomicAdd(&sm[OFF_EBAR + nD], ssum);
    }
  }
  __syncthreads();
  if (tid < DO_) out[(size_t)b * DO_ + tid] = sm[OFF_EBAR + tid];
}

extern "C" void kernel_launch(void* const* d_in, const int* in_sizes, int n_in,
                              void* d_out, int out_size, void* d_ws, size_t ws_size,
                              hipStream_t stream) {
  (void)n_in; (void)d_ws; (void)ws_size; (void)out_size;
  const float* x     = (const float*)d_in[0];
  const float* fr_w1 = (const float*)d_in[1];
  const float* fr_b1 = (const float*)d_in[2];
  const float* fr_w2 = (const float*)d_in[3];
  const float* fr_b2 = (const float*)d_in[4];
  const float* fr_w3 = (const float*)d_in[5];
  const float* fr_b3 = (const float*)d_in[6];
  const float* fo_w1 = (const float*)d_in[7];
  const float* fo_b1 = (const float*)d_in[8];
  const float* fo_w2 = (const float*)d_in[9];
  const float* fo_b2 = (const float*)d_in[10];
  const float* fo_w3 = (const float*)d_in[11];
  const float* fo_b3 = (const float*)d_in[12];
  float* out = (float*)d_out;

  const int batch = in_sizes[0] / (PFEAT * NPART);
  const size_t shmem = (size_t)LDS_FLOATS * sizeof(float);

  intnet_tagger_kernel<<<dim3(batch), dim3(256), shmem, stream>>>(
      x, fr_w1, fr_b1, fr_w2, fr_b2, fr_w3, fr_b3,
      fo_w1, fo_b1, fo_w2, fo_b2, fo_w3, fo_b3, out);
}